// StateModelEncoder_33560874451013
// MI455X (gfx1250) — compile-verified
//
#include <hip/hip_runtime.h>

#define DIN 5
#define HD  64

typedef __attribute__((ext_vector_type(2))) float v2f;
typedef __attribute__((ext_vector_type(8))) float v8f;

// ---------------------------------------------------------------------------
// Generic multi-term WMMA GEMM:  out[M,N] = relu?( sum_i (rs_i * A_i) @ W_i + bias (+bias2) )
// A_i : [M, K_i] row-major, W_i : [K_i, N] row-major, rs_i : optional per-row scale
// Uses V_WMMA_F32_16X16X4_F32 (full fp32). One wave computes one 16x16 tile of out.
// K and N are compile-time (K in {5,64}, N in {8,64}): the k-loop fully unrolls,
// B loads get immediate offsets (k*N is constant), and for N=64 the column
// masking folds away entirely.
// ---------------------------------------------------------------------------
struct Term { const float* A; const float* W; const float* rs; int K; };
struct GemmArgs {
  Term t[8];
  const float* bias;
  const float* bias2;
  float* out;
  int M, N, relu, nterms, accumulate;
};

template <int K, int N>
__device__ inline void gemm_term(v8f& acc, const float* __restrict__ A,
                                 const float* __restrict__ Wm,
                                 const float* __restrict__ rsp,
                                 int row, int half, int col, bool colok) {
  float rs = rsp ? rsp[row] : 1.0f;
  const float* Arow = A + (size_t)row * K;
  int colsafe = colok ? col : 0;
  const float* Wcol = Wm + colsafe;     // per-term base; k*N folds to immediate offsets
#pragma unroll
  for (int k0 = 0; k0 < K; k0 += 4) {
    int k = k0 + half * 2;              // half 0 -> (k0,k0+1), half 1 -> (k0+2,k0+3)
    v2f a, b;
    if ((K & 3) == 0) {
      a.x = Arow[k] * rs;               // merges to global_load_b64 (immediate offset)
      a.y = Arow[k + 1] * rs;
      float b0 = Wcol[k * N];
      float b1 = Wcol[(k + 1) * N];
      b.x = colok ? b0 : 0.f;           // folds away when colok is constant-true
      b.y = colok ? b1 : 0.f;
    } else {
      // branch-free zero padding: clamp address, mask value
      bool v0 = k < K, v1 = (k + 1) < K;
      int ka = v0 ? k : 0;
      int kb = v1 ? (k + 1) : 0;
      float a0 = Arow[ka];
      float a1 = Arow[kb];
      a.x = v0 ? a0 * rs : 0.f;
      a.y = v1 ? a1 * rs : 0.f;
      float b0 = Wcol[ka * N];
      float b1 = Wcol[kb * N];
      b.x = (v0 && colok) ? b0 : 0.f;
      b.y = (v1 && colok) ? b1 : 0.f;
    }
    acc = __builtin_amdgcn_wmma_f32_16x16x4_f32(false, a, false, b,
                                                (short)0, acc, false, false);
  }
}

template <int N>
__global__ __launch_bounds__(256) void wmma_gemm_kernel(GemmArgs g) {
  int gid  = blockIdx.x * blockDim.x + threadIdx.x;
  int wave = gid >> 5;
  int lane = threadIdx.x & 31;
  constexpr int ntiles = (N + 15) >> 4;
  int mt = wave / ntiles;               // ntiles is 4 or 1 -> shift/identity
  int nt = wave - mt * ntiles;
  int row0 = mt << 4;
  if (row0 >= g.M) return;              // wave-uniform exit (EXEC stays all-ones)
  int n0   = nt << 4;
  int half = lane >> 4;
  int mr   = lane & 15;
  int col  = n0 + mr;
  constexpr bool full = (N % 16) == 0;
  bool colok = full || (col < N);
  int arow = row0 + mr;                 // valid: all M here are multiples of 16

  v8f acc = {0.f, 0.f, 0.f, 0.f, 0.f, 0.f, 0.f, 0.f};

  for (int ti = 0; ti < g.nterms; ++ti) {
    const Term& t = g.t[ti];
    if (t.K == HD)
      gemm_term<HD, N>(acc, t.A, t.W, t.rs, arow, half, col, colok);
    else
      gemm_term<DIN, N>(acc, t.A, t.W, t.rs, arow, half, col, colok);
  }

  float bv = 0.f;
  if (g.bias  && colok) bv  = g.bias[col];
  if (g.bias2 && colok) bv += g.bias2[col];

#pragma unroll
  for (int v = 0; v < 8; ++v) {
    int row = row0 + v + half * 8;
    if (row < g.M && colok) {
      size_t o = (size_t)row * N + col;
      float val = acc[v] + bv;
      if (g.accumulate) val += g.out[o];
      if (g.relu) val = fmaxf(val, 0.f);
      g.out[o] = val;
    }
  }
}

// ---------------------------------------------------------------------------
// Edge kernels (scatter-side; atomics resolve in the 192MB L2)
// ---------------------------------------------------------------------------
__global__ void rgcn_agg_kernel(const int* __restrict__ ei, const int* __restrict__ et,
                                int E, int nv, const float* __restrict__ x,
                                float* agg, float* cnt) {
  int e = blockIdx.x * blockDim.x + threadIdx.x;
  if (e >= E) return;
  int s = ei[e], d = ei[E + e], r = et[e];
  float* ag = agg + ((size_t)r * nv + d) * DIN;
  const float* xs = x + (size_t)s * DIN;
#pragma unroll
  for (int j = 0; j < DIN; ++j) atomicAdd(ag + j, xs[j]);
  atomicAdd(cnt + (size_t)r * nv + d, 1.f);
}

__global__ void count_kernel(const int* __restrict__ ei, int E, float* cnt) {
  int e = blockIdx.x * blockDim.x + threadIdx.x;
  if (e >= E) return;
  atomicAdd(cnt + ei[E + e], 1.f);
}

__global__ void dinv_kernel(const float* __restrict__ deg, float* dinv, int n) {
  int i = blockIdx.x * blockDim.x + threadIdx.x;
  if (i >= n) return;
  float d = deg[i];
  dinv[i] = d > 0.f ? rsqrtf(fmaxf(d, 1.f)) : 0.f;
}

__global__ void rcp_kernel(const float* __restrict__ c, float* r, int n) {
  int i = blockIdx.x * blockDim.x + threadIdx.x;
  if (i >= n) return;
  r[i] = 1.f / fmaxf(c[i], 1.f);
}

// scatter-add of D-dim rows in CH-wide chunks: out[dst,j0..] += w * h[src,j0..]
// w = ew[e] and/or dinv[s]*dinv[d]. CH=4 gives b128 gather loads; edge-index
// and weight loads amortized over CH dims.
template <int D, int CH>
__global__ void scatter_kernel(const int* __restrict__ ei, int E,
                               const float* __restrict__ h,
                               const float* __restrict__ ew,
                               const float* __restrict__ dinv,
                               float* __restrict__ out) {
  constexpr int CPE = D / CH;           // chunks per edge
  long long idx = (long long)blockIdx.x * blockDim.x + threadIdx.x;
  if (idx >= (long long)E * CPE) return;
  int e  = (int)(idx / CPE);
  int j0 = (int)(idx - (long long)e * CPE) * CH;
  int s = ei[e], d = ei[E + e];
  float w = 1.f;
  if (ew)   w = ew[e];
  if (dinv) w *= dinv[s] * dinv[d];
  const float* hp = h + (size_t)s * D + j0;
  float* op = out + (size_t)d * D + j0;
  float v[CH];
#pragma unroll
  for (int c = 0; c < CH; ++c) v[c] = hp[c];
#pragma unroll
  for (int c = 0; c < CH; ++c) atomicAdd(op + c, w * v[c]);
}

// ---------------------------------------------------------------------------
// Host orchestration
// ---------------------------------------------------------------------------
static inline int cdiv(long long a, int b) { return (int)((a + b - 1) / b); }

static void launch_gemm(const GemmArgs& g, hipStream_t s) {
  int ntiles = (g.N + 15) >> 4;
  int mtiles = (g.M + 15) >> 4;
  long long threads = (long long)mtiles * ntiles * 32;
  int blocks = cdiv(threads, 256);
  if (g.N == HD)
    hipLaunchKernelGGL((wmma_gemm_kernel<HD>), dim3(blocks), dim3(256), 0, s, g);
  else
    hipLaunchKernelGGL((wmma_gemm_kernel<8>),  dim3(blocks), dim3(256), 0, s, g);
}

extern "C" void kernel_launch(void* const* d_in, const int* in_sizes, int n_in,
                              void* d_out, int out_size, void* d_ws, size_t ws_size,
                              hipStream_t stream) {
  const float* game_x  = (const float*)d_in[0];
  const float* state_x = (const float*)d_in[1];
  const int*   ei_vv   = (const int*)d_in[2];
  const int*   et_vv   = (const int*)d_in[3];
  const int*   ei_h    = (const int*)d_in[4];
  const float* ew_h    = (const float*)d_in[5];
  const int*   ei_in   = (const int*)d_in[6];
  const int*   ei_ss   = (const int*)d_in[7];
  const float *W1_rel = (const float*)d_in[8],  *W1_root = (const float*)d_in[9],  *b1  = (const float*)d_in[10];
  const float *W12    = (const float*)d_in[11], *b12     = (const float*)d_in[12];
  const float *W2     = (const float*)d_in[13], *b2      = (const float*)d_in[14];
  const float *W3_rel = (const float*)d_in[15], *W3_root = (const float*)d_in[16], *b3  = (const float*)d_in[17];
  const float *W32_l  = (const float*)d_in[18], *W32_r   = (const float*)d_in[19], *b32 = (const float*)d_in[20];
  const float *W4_l   = (const float*)d_in[21], *W4_r    = (const float*)d_in[22], *b4  = (const float*)d_in[23];
  const float *W42_l  = (const float*)d_in[24], *W42_r   = (const float*)d_in[25], *b42 = (const float*)d_in[26];
  const float *W5_l   = (const float*)d_in[27], *W5_r    = (const float*)d_in[28], *b5  = (const float*)d_in[29];
  const float *Wl     = (const float*)d_in[30], *bl      = (const float*)d_in[31];

  const int nv  = in_sizes[0] / DIN;        // 100000
  const int ns  = in_sizes[1] / DIN;        // 50000
  const int EVV = in_sizes[3];              // etype length
  const int EH  = in_sizes[5];              // edge_attr length
  const int EIN = in_sizes[6] / 2;
  const int ESS = in_sizes[7] / 2;

  // ---- workspace layout (floats; every block size is a multiple of 4 floats,
  //      so all 64-dim buffers are 16B aligned for b128 loads) ----
  float* ws = (float*)d_ws;
  size_t off = 0;
  auto take = [&](size_t n) { size_t o = off; off += n; return o; };
  const size_t A_RGCN = take((size_t)3 * nv * DIN);
  const size_t C_RGCN = take((size_t)3 * nv);
  const size_t R_RGCN = take((size_t)3 * nv);
  const size_t DEG_V  = take(nv);
  const size_t DINV_V = take(nv);
  const size_t H1 = take((size_t)nv * DIN);
  const size_t H2 = take((size_t)nv * DIN);
  const size_t H3 = take((size_t)nv * DIN);
  const size_t GX     = take((size_t)nv * HD);
  const size_t AGG_HW = take((size_t)ns * HD);
  const size_t AGG_H  = take((size_t)ns * HD);
  const size_t AGG_IN = take((size_t)ns * HD);
  const size_t CNT_H  = take(ns);
  const size_t RC_H   = take(ns);
  const size_t CNT_IN = take(ns);
  const size_t RC_IN  = take(ns);
  const size_t DEG_S  = take(ns);
  const size_t DINV_S = take(ns);
  const size_t RC_SS  = take(ns);
  const size_t SXA = take((size_t)ns * HD);
  const size_t SXB = take((size_t)ns * HD);
  const size_t total = off;
  // aliases: hop buffers for TAG-ss reuse dead regions (gx, hist aggs); agg_ss reuses agg_in
  const size_t HS1 = AGG_HW, HS2 = AGG_H, HS3 = GX, AGG_SS = AGG_IN;
  auto p = [&](size_t o) { return ws + o; };

  hipMemsetAsync(ws, 0, total * sizeof(float), stream);

  // ---- game graph: RGCN aggregation (5-dim), TAG hops (5-dim) ----
  hipLaunchKernelGGL(rgcn_agg_kernel, dim3(cdiv(EVV, 256)), dim3(256), 0, stream,
                     ei_vv, et_vv, EVV, nv, game_x, p(A_RGCN), p(C_RGCN));
  hipLaunchKernelGGL(count_kernel, dim3(cdiv(EVV, 256)), dim3(256), 0, stream, ei_vv, EVV, p(DEG_V));
  hipLaunchKernelGGL(dinv_kernel, dim3(cdiv(nv, 256)), dim3(256), 0, stream, p(DEG_V), p(DINV_V), nv);
  hipLaunchKernelGGL(rcp_kernel, dim3(cdiv(3 * nv, 256)), dim3(256), 0, stream, p(C_RGCN), p(R_RGCN), 3 * nv);
  hipLaunchKernelGGL((scatter_kernel<DIN, DIN>), dim3(cdiv(EVV, 256)), dim3(256), 0, stream,
                     ei_vv, EVV, game_x, (const float*)nullptr, p(DINV_V), p(H1));
  hipLaunchKernelGGL((scatter_kernel<DIN, DIN>), dim3(cdiv(EVV, 256)), dim3(256), 0, stream,
                     ei_vv, EVV, p(H1), (const float*)nullptr, p(DINV_V), p(H2));
  hipLaunchKernelGGL((scatter_kernel<DIN, DIN>), dim3(cdiv(EVV, 256)), dim3(256), 0, stream,
                     ei_vv, EVV, p(H2), (const float*)nullptr, p(DINV_V), p(H3));

  // gx = x@W1_root + sum_r mean_r@W1_rel[r] + b1 + x@W12[0] + sum_k hk@W12[k] + b12
  {
    GemmArgs g{};
    g.t[0] = {game_x,                  W1_root,        nullptr,        DIN};
    g.t[1] = {p(A_RGCN),               W1_rel,         p(R_RGCN),      DIN};
    g.t[2] = {p(A_RGCN) + (size_t)nv*DIN,   W1_rel + DIN*HD,   p(R_RGCN) + nv,     DIN};
    g.t[3] = {p(A_RGCN) + (size_t)2*nv*DIN, W1_rel + 2*DIN*HD, p(R_RGCN) + 2*nv,   DIN};
    g.t[4] = {game_x, W12,            nullptr, DIN};
    g.t[5] = {p(H1),  W12 + DIN*HD,   nullptr, DIN};
    g.t[6] = {p(H2),  W12 + 2*DIN*HD, nullptr, DIN};
    g.t[7] = {p(H3),  W12 + 3*DIN*HD, nullptr, DIN};
    g.nterms = 8; g.bias = b1; g.bias2 = b12; g.out = p(GX); g.M = nv; g.N = HD;
    launch_gemm(g, stream);
  }

  // ---- history edges: weighted sum (GraphConv) + plain sum & count (SAGE mean) ----
  hipLaunchKernelGGL((scatter_kernel<HD, 4>), dim3(cdiv((long long)EH * (HD / 4), 256)), dim3(256), 0, stream,
                     ei_h, EH, p(GX), ew_h, (const float*)nullptr, p(AGG_HW));
  hipLaunchKernelGGL((scatter_kernel<HD, 4>), dim3(cdiv((long long)EH * (HD / 4), 256)), dim3(256), 0, stream,
                     ei_h, EH, p(GX), (const float*)nullptr, (const float*)nullptr, p(AGG_H));
  hipLaunchKernelGGL(count_kernel, dim3(cdiv(EH, 256)), dim3(256), 0, stream, ei_h, EH, p(CNT_H));
  hipLaunchKernelGGL(rcp_kernel, dim3(cdiv(ns, 256)), dim3(256), 0, stream, p(CNT_H), p(RC_H), ns);

  { // sxA = relu(GraphConv)
    GemmArgs g{};
    g.t[0] = {p(AGG_HW), W3_rel,  nullptr, HD};
    g.t[1] = {state_x,   W3_root, nullptr, DIN};
    g.nterms = 2; g.bias = b3; g.out = p(SXA); g.M = ns; g.N = HD; g.relu = 1;
    launch_gemm(g, stream);
  }
  { // sxB = relu(SAGE32)
    GemmArgs g{};
    g.t[0] = {p(AGG_H), W32_l, p(RC_H), HD};
    g.t[1] = {p(SXA),   W32_r, nullptr, HD};
    g.nterms = 2; g.bias = b32; g.out = p(SXB); g.M = ns; g.N = HD; g.relu = 1;
    launch_gemm(g, stream);
  }

  // ---- "in" edges: one aggregation shared by both SAGE layers ----
  hipLaunchKernelGGL((scatter_kernel<HD, 4>), dim3(cdiv((long long)EIN * (HD / 4), 256)), dim3(256), 0, stream,
                     ei_in, EIN, p(GX), (const float*)nullptr, (const float*)nullptr, p(AGG_IN));
  hipLaunchKernelGGL(count_kernel, dim3(cdiv(EIN, 256)), dim3(256), 0, stream, ei_in, EIN, p(CNT_IN));
  hipLaunchKernelGGL(rcp_kernel, dim3(cdiv(ns, 256)), dim3(256), 0, stream, p(CNT_IN), p(RC_IN), ns);

  { // sxA = relu(SAGE4)
    GemmArgs g{};
    g.t[0] = {p(AGG_IN), W4_l, p(RC_IN), HD};
    g.t[1] = {p(SXB),    W4_r, nullptr,  HD};
    g.nterms = 2; g.bias = b4; g.out = p(SXA); g.M = ns; g.N = HD; g.relu = 1;
    launch_gemm(g, stream);
  }
  { // sxB = relu(SAGE42)  (same mean aggregation of gx)
    GemmArgs g{};
    g.t[0] = {p(AGG_IN), W42_l, p(RC_IN), HD};
    g.t[1] = {p(SXA),    W42_r, nullptr,  HD};
    g.nterms = 2; g.bias = b42; g.out = p(SXB); g.M = ns; g.N = HD; g.relu = 1;
    launch_gemm(g, stream);
  }

  // ---- s_s graph: TAG(H, K=3) then SAGE ----
  hipLaunchKernelGGL(count_kernel, dim3(cdiv(ESS, 256)), dim3(256), 0, stream, ei_ss, ESS, p(DEG_S));
  hipLaunchKernelGGL(dinv_kernel, dim3(cdiv(ns, 256)), dim3(256), 0, stream, p(DEG_S), p(DINV_S), ns);
  hipLaunchKernelGGL(rcp_kernel, dim3(cdiv(ns, 256)), dim3(256), 0, stream, p(DEG_S), p(RC_SS), ns);

  // zero aliased hop buffers (AGG_HW/AGG_H contiguous; GX region separately)
  hipMemsetAsync(p(HS1), 0, (size_t)2 * ns * HD * sizeof(float), stream);
  hipMemsetAsync(p(HS3), 0, (size_t)ns * HD * sizeof(float), stream);
  hipLaunchKernelGGL((scatter_kernel<HD, 4>), dim3(cdiv((long long)ESS * (HD / 4), 256)), dim3(256), 0, stream,
                     ei_ss, ESS, p(SXB), (const float*)nullptr, p(DINV_S), p(HS1));
  hipLaunchKernelGGL((scatter_kernel<HD, 4>), dim3(cdiv((long long)ESS * (HD / 4), 256)), dim3(256), 0, stream,
                     ei_ss, ESS, p(HS1), (const float*)nullptr, p(DINV_S), p(HS2));
  hipLaunchKernelGGL((scatter_kernel<HD, 4>), dim3(cdiv((long long)ESS * (HD / 4), 256)), dim3(256), 0, stream,
                     ei_ss, ESS, p(HS2), (const float*)nullptr, p(DINV_S), p(HS3));

  { // sxA = relu(TAG)
    GemmArgs g{};
    g.t[0] = {p(SXB), W2,             nullptr, HD};
    g.t[1] = {p(HS1), W2 + HD*HD,     nullptr, HD};
    g.t[2] = {p(HS2), W2 + 2*HD*HD,   nullptr, HD};
    g.t[3] = {p(HS3), W2 + 3*HD*HD,   nullptr, HD};
    g.nterms = 4; g.bias = b2; g.out = p(SXA); g.M = ns; g.N = HD; g.relu = 1;
    launch_gemm(g, stream);
  }

  // SAGE5 over s_s
  hipMemsetAsync(p(AGG_SS), 0, (size_t)ns * HD * sizeof(float), stream);
  hipLaunchKernelGGL((scatter_kernel<HD, 4>), dim3(cdiv((long long)ESS * (HD / 4), 256)), dim3(256), 0, stream,
                     ei_ss, ESS, p(SXA), (const float*)nullptr, (const float*)nullptr, p(AGG_SS));
  { // sxB = relu(SAGE5)
    GemmArgs g{};
    g.t[0] = {p(AGG_SS), W5_l, p(RC_SS), HD};
    g.t[1] = {p(SXA),    W5_r, nullptr,  HD};
    g.nterms = 2; g.bias = b5; g.out = p(SXB); g.M = ns; g.N = HD; g.relu = 1;
    launch_gemm(g, stream);
  }

  { // final linear: out = sxB @ Wl + bl   [NS, 8]
    GemmArgs g{};
    g.t[0] = {p(SXB), Wl, nullptr, HD};
    g.nterms = 1; g.bias = bl; g.out = (float*)d_out; g.M = ns; g.N = 8;
    launch_gemm(g, stream);
  }
}